// SpatialSimilarityFeatures_53841710023043
// MI455X (gfx1250) — compile-verified
//
#include <hip/hip_runtime.h>

// SpatialSimilarityFeatures fused kernel for gfx1250 (MI455X).
// One workgroup (256 threads = 8 wave32) per (B,Wn) window.
// Phase 1: per-pixel L1 similarity vs all 256 pixels + top-16 (registers).
// Phase 2: feature build (scores + rel grid pos) into LDS.
// Phase 3: six GEMM layers via V_WMMA_F32_16X16X4_F32, wave-private 16-row tiles.
//          Weights staged into LDS in transposed 32-column panels so B-fragments
//          are conflict-free ds_load_b64 (no global-latency hoisting -> no spills).

#define NWIN 64
#define PIX  256
#define CH   32
#define KNN  16
#define ASTR 132   // LDS activation row stride (floats); 132 % 64 = 4 -> conflict-free WMMA A/D access

typedef __attribute__((ext_vector_type(2))) float v2f;
typedef __attribute__((ext_vector_type(8))) float v8f;

// One WMMA layer for rows [r0, r0+16): act[cols inCol..inCol+KD) x W[KD][NO] + bias,
// optional ReLU; result to act cols [outCol, outCol+NO) or straight to global (GLB).
// All A fragments preloaded to registers first => in-place (inCol==outCol) is safe.
// Weight panels (32 output cols, transposed) staged cooperatively in LDS; the two
// __syncthreads per panel keep all 8 waves in lockstep (uniform control flow).
template<int KD, int NO, bool RELU, bool GLB>
__device__ __forceinline__ void ffn_layer(float* act, float* wpanel,
                                          int r0, int inCol, int outCol,
                                          const float* __restrict__ W,
                                          const float* __restrict__ bias,
                                          float* __restrict__ gout,
                                          int tid)
{
    constexpr int KS   = KD / 4;
    constexpr int KSTR = KD + 4;          // transposed-panel row stride (banks offset 4/row)
    const int lane = tid & 31, li = lane & 15, lh = lane >> 4;

    // Preload this wave's A fragments (16 rows x KD) -- makes in-place update safe.
    v2f a[KS];
    {
        const float* inrow = act + (r0 + li) * ASTR + inCol + 2 * lh;
        #pragma unroll
        for (int k = 0; k < KS; ++k) { a[k].x = inrow[4 * k]; a[k].y = inrow[4 * k + 1]; }
    }

    #pragma clang loop unroll(disable)
    for (int np = 0; np < NO / 32; ++np) {
        __syncthreads();   // previous panel fully consumed by all waves
        // Cooperative stage: wpanel[nc][k] = W[k][np*32+nc]   (32*KD multiple of 256)
        for (int e = tid; e < 32 * KD; e += 256) {
            int nc = e & 31, k = e >> 5;
            wpanel[nc * KSTR + k] = W[k * NO + np * 32 + nc];
        }
        __syncthreads();

        #pragma unroll
        for (int nn = 0; nn < 2; ++nn) {
            const int n = np * 2 + nn;
            float bv = bias[n * 16 + li];
            v8f acc = {bv, bv, bv, bv, bv, bv, bv, bv};
            const float* wp = wpanel + (nn * 16 + li) * KSTR + 2 * lh;
            #pragma unroll
            for (int k = 0; k < KS; ++k) {
                v2f b;
                b.x = wp[4 * k];          // contiguous pair -> ds_load_b64
                b.y = wp[4 * k + 1];
                acc = __builtin_amdgcn_wmma_f32_16x16x4_f32(false, a[k], false, b,
                                                            (short)0, acc, false, false);
            }
            #pragma unroll
            for (int v = 0; v < 8; ++v) {
                float d = acc[v];
                if (RELU) d = fmaxf(d, 0.0f);
                if (GLB)  gout[(size_t)(r0 + v + 8 * lh) * 64 + n * 16 + li] = d;
                else      act[(r0 + v + 8 * lh) * ASTR + outCol + n * 16 + li] = d;
            }
        }
    }
}

__global__ __launch_bounds__(256)
void ssf_kernel(const float* __restrict__ x,
                const float* __restrict__ sf_w0, const float* __restrict__ sf_b0,
                const float* __restrict__ sf_w1, const float* __restrict__ sf_b1,
                const float* __restrict__ sf_wo, const float* __restrict__ sf_bo,
                const float* __restrict__ out_w0, const float* __restrict__ out_b0,
                const float* __restrict__ out_w1, const float* __restrict__ out_b1,
                const float* __restrict__ out_wo, const float* __restrict__ out_bo,
                float* __restrict__ out)
{
    __shared__ float act[PIX * ASTR];    // 135 KB activation scratch
    __shared__ float wpanel[32 * 132];   // 16.5 KB transposed weight panel

    const int tid = threadIdx.x;
    const int win = blockIdx.x;
    const float* __restrict__ xw = x + (size_t)win * PIX * CH;

    // Warm L2/L1 for the (shared, tiny) weight matrices.
    __builtin_prefetch(sf_w0 + tid * 24, 0, 1);   // 48x128
    __builtin_prefetch(sf_w1 + tid * 64, 0, 1);   // 128x128
    __builtin_prefetch(sf_wo + tid * 32, 0, 1);   // 128x64
    __builtin_prefetch(out_w0 + tid * 48, 0, 1);  // 96x128
    __builtin_prefetch(out_w1 + tid * 64, 0, 1);  // 128x128
    __builtin_prefetch(out_wo + tid * 32, 0, 1);  // 128x64

    // ---- Phase 1: L1 similarity + stable top-16 (registers) -----------------
    float4 xi4[8];
    #pragma unroll
    for (int c = 0; c < 8; ++c) xi4[c] = ((const float4*)(xw + tid * CH))[c];

    float sc[KNN]; int id[KNN];
    #pragma unroll
    for (int t = 0; t < KNN; ++t) { sc[t] = -3.0e38f; id[t] = 0; }

    for (int j = 0; j < PIX; ++j) {
        const float4* __restrict__ xr = (const float4*)(xw + j * CH); // uniform addr -> broadcast
        float s0 = 0.f, s1 = 0.f, s2 = 0.f, s3 = 0.f;
        #pragma unroll
        for (int c = 0; c < 8; ++c) {
            float4 v = xr[c];
            s0 += fabsf(xi4[c].x - v.x);
            s1 += fabsf(xi4[c].y - v.y);
            s2 += fabsf(xi4[c].z - v.z);
            s3 += fabsf(xi4[c].w - v.w);
        }
        float s = 1.0f - (s0 + s1 + s2 + s3) * (1.0f / 32.0f);
        if (s > sc[KNN - 1]) {            // strict '>' keeps jax tie order (lower index first)
            sc[KNN - 1] = s; id[KNN - 1] = j;
            #pragma unroll
            for (int t = KNN - 1; t > 0; --t) {
                if (sc[t] > sc[t - 1]) {
                    float ts = sc[t]; sc[t] = sc[t - 1]; sc[t - 1] = ts;
                    int ti = id[t];  id[t] = id[t - 1];  id[t - 1] = ti;
                }
            }
        }
    }

    // ---- Phase 2: features [scores(16) | rel_pos(32)] into LDS --------------
    {
        float hp = (float)(tid >> 4), wq = (float)(tid & 15);
        float* arow = act + tid * ASTR;
        #pragma unroll
        for (int k = 0; k < KNN; ++k) {
            arow[k] = sc[k];
            int q = id[k];
            arow[16 + 2 * k]     = (hp - (float)(q >> 4)) * (1.0f / 15.0f);
            arow[16 + 2 * k + 1] = (wq - (float)(q & 15)) * (1.0f / 15.0f);
        }
    }
    __syncthreads();

    // ---- Phase 3: WMMA FFNs, full 6-layer chain per wave-private 16-row block
    const int wave = tid >> 5, lane = tid & 31, li = lane & 15, lh = lane >> 4;
    float* __restrict__ ow = out + (size_t)win * PIX * 64;

    #pragma clang loop unroll(disable)
    for (int m = 0; m < 2; ++m) {
        const int r0 = wave * 32 + m * 16;
        ffn_layer<48, 128, true,  false>(act, wpanel, r0, 0, 0,  sf_w0, sf_b0, nullptr, tid);
        ffn_layer<128, 128, true, false>(act, wpanel, r0, 0, 0,  sf_w1, sf_b1, nullptr, tid);
        ffn_layer<128, 64, false, false>(act, wpanel, r0, 0, 32, sf_wo, sf_bo, nullptr, tid); // sf -> cols 32..95
        // Concat: x -> cols 0..31 of this block's rows (lane L: row r0+(L&15), cols (L>>4)*16..+15).
        {
            const int row = r0 + li, cb = lh * 16;
            const float4* __restrict__ src = (const float4*)(xw + row * CH + cb);
            float4* dst = (float4*)(act + row * ASTR + cb);
            #pragma unroll
            for (int c = 0; c < 4; ++c) dst[c] = src[c];   // in-wave LDS order: safe
        }
        ffn_layer<96, 128, true,  false>(act, wpanel, r0, 0, 0, out_w0, out_b0, nullptr, tid);
        ffn_layer<128, 128, true, false>(act, wpanel, r0, 0, 0, out_w1, out_b1, nullptr, tid);
        ffn_layer<128, 64, false, true >(act, wpanel, r0, 0, 0, out_wo, out_bo, ow, tid);
    }
}

extern "C" void kernel_launch(void* const* d_in, const int* in_sizes, int n_in,
                              void* d_out, int out_size, void* d_ws, size_t ws_size,
                              hipStream_t stream) {
    (void)in_sizes; (void)n_in; (void)d_ws; (void)ws_size; (void)out_size;
    const float* x      = (const float*)d_in[0];
    const float* sf_w0  = (const float*)d_in[1];
    const float* sf_b0  = (const float*)d_in[2];
    const float* sf_w1  = (const float*)d_in[3];
    const float* sf_b1  = (const float*)d_in[4];
    const float* sf_wo  = (const float*)d_in[5];
    const float* sf_bo  = (const float*)d_in[6];
    const float* out_w0 = (const float*)d_in[7];
    const float* out_b0 = (const float*)d_in[8];
    const float* out_w1 = (const float*)d_in[9];
    const float* out_b1 = (const float*)d_in[10];
    const float* out_wo = (const float*)d_in[11];
    const float* out_bo = (const float*)d_in[12];

    ssf_kernel<<<NWIN, 256, 0, stream>>>(x, sf_w0, sf_b0, sf_w1, sf_b1, sf_wo, sf_bo,
                                         out_w0, out_b0, out_w1, out_b1, out_wo, out_bo,
                                         (float*)d_out);
}